// NashSuruKayipFonksiyonu_30030411334249
// MI455X (gfx1250) — compile-verified
//
#include <hip/hip_runtime.h>
#include <math.h>

typedef __attribute__((ext_vector_type(2))) float v2f;
typedef __attribute__((ext_vector_type(8))) float v8f;

#define EPSF 1e-8f
#define NEG_BIG -3.402823466e38f

// ---------------------------------------------------------------------------
// ws layout (floats): [0]=ce_sum [1]=yuk_sum [2]=yuk_sq [3]=kaz_sum [4]=kaz_sq
//                     [5]=cpu_sum [6]=cos_sum [7]=valid_cnt
// ---------------------------------------------------------------------------
__global__ void init_ws(float* ws) {
    if (threadIdx.x < 8) ws[threadIdx.x] = 0.0f;
}

// ---------------------------------------------------------------------------
// Cross-entropy: one block (256 thr = 8 waves) per row, single-pass online
// softmax over V logits (float4 loads -> global_load_b128). Wave partials are
// combined with one exact f32 WMMA (A = ones(16x4), B columns 0/1 hold the 8
// rescaled partial sums -> D[0,0]+D[0,1] = total).
// ---------------------------------------------------------------------------
__global__ void ce_kernel(const float* __restrict__ logits,
                          const int* __restrict__ tgt,
                          int V, float* __restrict__ acc) {
    const int row = blockIdx.x;
    const float* rp = logits + (size_t)row * (size_t)V;
    const float4* rp4 = (const float4*)rp;
    const int nvec = V >> 2;

    float m = NEG_BIG, s = 0.0f;
    for (int i = threadIdx.x; i < nvec; i += blockDim.x) {
        float4 x = rp4[i];
        float mx = fmaxf(fmaxf(x.x, x.y), fmaxf(x.z, x.w));
        if (mx > m) { s *= __expf(m - mx); m = mx; }
        s += __expf(x.x - m) + __expf(x.y - m) + __expf(x.z - m) + __expf(x.w - m);
    }
    // wave32 (m,s) combine
    #pragma unroll
    for (int o = 16; o > 0; o >>= 1) {
        float mo = __shfl_xor(m, o, 32);
        float so = __shfl_xor(s, o, 32);
        float M = fmaxf(m, mo);
        s = s * __expf(m - M) + so * __expf(mo - M);
        m = M;
    }
    __shared__ float sm[8], ss[8];
    const int wid = threadIdx.x >> 5, lid = threadIdx.x & 31;
    if (lid == 0) { sm[wid] = m; ss[wid] = s; }
    __syncthreads();

    if (threadIdx.x < 32) {   // wave 0, full EXEC (WMMA requirement)
        float M = sm[0];
        #pragma unroll
        for (int i = 1; i < 8; ++i) M = fmaxf(M, sm[i]);
        const int l = threadIdx.x;
        // B (4x16 f32, 2 VGPRs): v0 lanes0-15=(K0,N=lane), lanes16-31=(K2,N=lane-16)
        //                        v1 lanes0-15=(K1,N=lane), lanes16-31=(K3,N=lane-16)
        // Put partial i at B[i&3, i>>2]  (columns 0 and 1)
        const bool act = (l < 2) | (l == 16) | (l == 17);
        const int base = ((l & 1) << 2) + ((l >> 4) << 1);
        float b0 = act ? ss[base]     * __expf(sm[base]     - M) : 0.0f;
        float b1 = act ? ss[base + 1] * __expf(sm[base + 1] - M) : 0.0f;
        v2f a; a[0] = 1.0f; a[1] = 1.0f;          // A = ones(16x4)
        v2f b; b[0] = b0;   b[1] = b1;
        v8f c = {};
        v8f d = __builtin_amdgcn_wmma_f32_16x16x4_f32(
            false, a, false, b, (short)0, c, false, false);
        // D[0,0] in lane0.v0, D[0,1] in lane1.v0
        float S = d[0] + __shfl(d[0], 1, 32);
        if (l == 0) {
            const int t = tgt[row];
            const float xt = rp[t];
            atomicAdd(acc, M + __logf(S) - xt);   // -log p(target)
        }
    }
}

// ---------------------------------------------------------------------------
// sum and sum-of-squares reduction (for var/mean terms)
// ---------------------------------------------------------------------------
__global__ void sumsq_kernel(const float* __restrict__ x, int n,
                             float* __restrict__ acc /* [0]=sum [1]=sumsq */) {
    float s = 0.0f, s2 = 0.0f;
    for (int i = blockIdx.x * blockDim.x + threadIdx.x; i < n;
         i += gridDim.x * blockDim.x) {
        float v = x[i];
        s += v; s2 += v * v;
    }
    #pragma unroll
    for (int o = 16; o > 0; o >>= 1) {
        s  += __shfl_xor(s,  o, 32);
        s2 += __shfl_xor(s2, o, 32);
    }
    __shared__ float as[8], as2[8];
    const int wid = threadIdx.x >> 5, lid = threadIdx.x & 31;
    if (lid == 0) { as[wid] = s; as2[wid] = s2; }
    __syncthreads();
    if (threadIdx.x == 0) {
        float S = 0.0f, S2 = 0.0f;
        #pragma unroll
        for (int i = 0; i < 8; ++i) { S += as[i]; S2 += as2[i]; }
        atomicAdd(&acc[0], S);
        atomicAdd(&acc[1], S2);
    }
}

// ---------------------------------------------------------------------------
// mean of cache[sel[i]]
// ---------------------------------------------------------------------------
__global__ void gather_mean_kernel(const float* __restrict__ cache,
                                   const int* __restrict__ sel, int n,
                                   float* __restrict__ acc) {
    float s = 0.0f;
    for (int i = blockIdx.x * blockDim.x + threadIdx.x; i < n;
         i += gridDim.x * blockDim.x)
        s += cache[sel[i]];
    #pragma unroll
    for (int o = 16; o > 0; o >>= 1) s += __shfl_xor(s, o, 32);
    __shared__ float as[8];
    const int wid = threadIdx.x >> 5, lid = threadIdx.x & 31;
    if (lid == 0) as[wid] = s;
    __syncthreads();
    if (threadIdx.x == 0) {
        float S = 0.0f;
        #pragma unroll
        for (int i = 0; i < 8; ++i) S += as[i];
        atomicAdd(acc, S);
    }
}

// ---------------------------------------------------------------------------
// Swarm cohesion: one wave32 per row (D=64 -> one float2 per lane). Neighbor
// rows gathered with b64 loads; strat fits in the 192 MB L2 so gathers hit
// cache. global_prefetch on the next neighbor row.
// ---------------------------------------------------------------------------
__global__ void suru_kernel(const float* __restrict__ strat,
                            const int* __restrict__ komsu,
                            int n, int K,
                            float* __restrict__ acc /* [0]=cos_sum [1]=cnt */) {
    const int lane = threadIdx.x & 31;
    const int row = blockIdx.x * (blockDim.x >> 5) + (threadIdx.x >> 5);
    if (row >= n) return;

    const float2* sp = (const float2*)strat;       // row r = float2 index r*32
    const float2 s = sp[(size_t)row * 32 + lane];
    const int* kr = komsu + (size_t)row * K;

    float ax = 0.0f, ay = 0.0f;
    for (int k = 0; k < K; ++k) {
        const int idx = kr[k];
        if (k + 1 < K)
            __builtin_prefetch(&sp[(size_t)kr[k + 1] * 32], 0, 0);
        const float2 nb = sp[(size_t)idx * 32 + lane];
        ax += nb.x; ay += nb.y;
    }
    const float invK = 1.0f / (float)K;
    ax *= invK; ay *= invK;

    float dot = s.x * ax + s.y * ay;
    float bn2 = s.x * s.x + s.y * s.y;
    float kn2 = ax * ax + ay * ay;
    #pragma unroll
    for (int o = 16; o > 0; o >>= 1) {
        dot += __shfl_xor(dot, o, 32);
        bn2 += __shfl_xor(bn2, o, 32);
        kn2 += __shfl_xor(kn2, o, 32);
    }
    if (lane == 0) {
        const float bn = sqrtf(bn2), kn = sqrtf(kn2);
        const bool valid = (bn > EPSF) && (kn > EPSF);
        const float cosv = valid ? dot / (bn * kn) : 0.0f;
        atomicAdd(&acc[0], cosv);
        atomicAdd(&acc[1], valid ? 1.0f : 0.0f);
    }
}

// ---------------------------------------------------------------------------
// finalize -> 6 output floats
// ---------------------------------------------------------------------------
__global__ void finalize_kernel(const float* __restrict__ ws,
                                float* __restrict__ out,
                                int n_tok, int n_birim) {
    if (threadIdx.x == 0 && blockIdx.x == 0) {
        const float nt = (float)n_tok, nb = (float)n_birim;
        const float temel = ws[0] / nt;

        const float m1 = ws[1] / nb;
        const float v1 = (ws[2] - nb * m1 * m1) / (nb - 1.0f);
        const float denge = v1 / (m1 + EPSF);

        const float cpu = ws[5] / nt;

        const float cnt = ws[7];
        const float uyum = (cnt > 0.0f) ? (ws[6] / cnt) : 0.0f;
        const float suru = (uyum + 1.0f) * 0.5f;

        const float m2 = ws[3] / nb;
        const float v2 = (ws[4] - nb * m2 * m2) / (nb - 1.0f);
        const float nash = v2 / (m2 + EPSF);

        out[0] = 1.0f * temel + 0.1f * denge - 0.05f * cpu
                 - 0.03f * suru + 0.02f * nash;
        out[1] = temel;
        out[2] = denge;
        out[3] = cpu;
        out[4] = suru;
        out[5] = nash;
    }
}

// ---------------------------------------------------------------------------
extern "C" void kernel_launch(void* const* d_in, const int* in_sizes, int n_in,
                              void* d_out, int out_size, void* d_ws, size_t ws_size,
                              hipStream_t stream) {
    const float* logits = (const float*)d_in[0];   // [N_tok, V]
    const int*   tgt    = (const int*)  d_in[1];   // [N_tok]
    const float* yuk    = (const float*)d_in[2];   // [N_birim]
    const float* onb    = (const float*)d_in[3];   // [N_birim]
    const int*   sel    = (const int*)  d_in[4];   // [N_tok]
    const float* strat  = (const float*)d_in[5];   // [N_birim, 64]
    const int*   komsu  = (const int*)  d_in[6];   // [N_birim, K]
    const float* kaz    = (const float*)d_in[7];   // [N_birim]
    // d_in[8] (mevcut_stratejiler) unused, as in the reference.

    const int n_tok   = in_sizes[1];
    const int V       = in_sizes[0] / n_tok;
    const int n_birim = in_sizes[2];
    const int K       = in_sizes[6] / n_birim;

    float* ws  = (float*)d_ws;
    float* out = (float*)d_out;

    init_ws<<<1, 32, 0, stream>>>(ws);
    ce_kernel<<<n_tok, 256, 0, stream>>>(logits, tgt, V, ws + 0);
    sumsq_kernel<<<256, 256, 0, stream>>>(yuk, n_birim, ws + 1);
    sumsq_kernel<<<256, 256, 0, stream>>>(kaz, n_birim, ws + 3);
    gather_mean_kernel<<<32, 256, 0, stream>>>(onb, sel, n_tok, ws + 5);
    suru_kernel<<<(n_birim + 7) / 8, 256, 0, stream>>>(strat, komsu, n_birim, K, ws + 6);
    finalize_kernel<<<1, 1, 0, stream>>>(ws, out, n_tok, n_birim);
}